// GNNEncoder_10496900071608
// MI455X (gfx1250) — compile-verified
//
#include <hip/hip_runtime.h>
#include <math.h>

#define NNODES 100000
#define NEDGES 1200000
#define FDIM   64
#define ROWTILES (NNODES / 16)   // 6250, exact

typedef __attribute__((ext_vector_type(2))) float v2f;
typedef __attribute__((ext_vector_type(8))) float v8f;

// ---------------------------------------------------------------------------
// Edge scatter: agg[dst] += h[src].  16 lanes per edge, float4 gather (256B
// per edge row, coalesced), 4x global_atomic_add_f32 per lane. agg fits in L2
// (25.6MB << 192MB) so atomics resolve at L2 rate.
// ---------------------------------------------------------------------------
__global__ void __launch_bounds__(256) scatter_add_kernel(
    const float* __restrict__ h, const int* __restrict__ src,
    const int* __restrict__ dst, float* __restrict__ agg, int nEdges)
{
    int t = blockIdx.x * blockDim.x + threadIdx.x;
    int e = t >> 4;                 // 16 threads per edge
    if (e >= nEdges) return;
    int q = t & 15;                 // float4 slot 0..15 (covers 64 floats)

    int s = src[e];
    int d = dst[e];

    float4 v = reinterpret_cast<const float4*>(h + (size_t)s * FDIM)[q];
    float* op = agg + (size_t)d * FDIM + q * 4;
    atomicAdd(op + 0, v.x);
    atomicAdd(op + 1, v.y);
    atomicAdd(op + 2, v.z);
    atomicAdd(op + 3, v.w);
}

// ---------------------------------------------------------------------------
// Fused SAGE dense stage: out = act(agg @ Wl^T + bl + hin @ Wr^T)
// One wave computes a 16x64 output slab with V_WMMA_F32_16X16X4_F32,
// accumulating both GEMMs into the same 4 v8f accumulators over K=64
// (16 steps of K=4). Weights staged in LDS (2 x 16KB).
//
// A-fragment (16x4 f32): lane<16 -> row=lane, {K=kb,kb+1}; lane>=16 ->
// row=lane-16, {K=kb+2,kb+3}  => contiguous float2 at h[row*64 + kb + half*2].
// B-fragment (4x16 f32): B[k][n] = W[n][k]  => contiguous float2 at
// W[n*64 + kb + half*2], n = n0 + (lane&15).
// ---------------------------------------------------------------------------
__global__ void __launch_bounds__(256) sage_gemm_kernel(
    const float* __restrict__ agg, const float* __restrict__ hin,
    const float* __restrict__ Wl,  const float* __restrict__ bias,
    const float* __restrict__ Wr,  float* __restrict__ out, int applyTanh)
{
    __shared__ float sWl[FDIM * FDIM];
    __shared__ float sWr[FDIM * FDIM];

    // Cooperative weight staging: 2 x 4096 floats with float4 moves.
    for (int i = threadIdx.x; i < (FDIM * FDIM) / 4; i += 256) {
        reinterpret_cast<float4*>(sWl)[i] = reinterpret_cast<const float4*>(Wl)[i];
        reinterpret_cast<float4*>(sWr)[i] = reinterpret_cast<const float4*>(Wr)[i];
    }
    __syncthreads();

    const int wave = threadIdx.x >> 5;   // 8 waves / block (wave32)
    const int lane = threadIdx.x & 31;
    const int half = lane >> 4;          // 0: lanes 0-15, 1: lanes 16-31
    const int lrow = lane & 15;

    const int rb = blockIdx.x * 8 + wave;
    if (rb >= ROWTILES) return;          // wave-uniform guard, after barrier
    const size_t r0 = (size_t)rb * 16;

    v8f acc[4];
    const v8f zero = {0.f, 0.f, 0.f, 0.f, 0.f, 0.f, 0.f, 0.f};
#pragma unroll
    for (int t = 0; t < 4; ++t) acc[t] = zero;

    const float* aRowL = agg + (r0 + lrow) * FDIM + half * 2;
    const float* aRowR = hin + (r0 + lrow) * FDIM + half * 2;

#pragma unroll
    for (int kb = 0; kb < FDIM; kb += 4) {
        v2f aL = *reinterpret_cast<const v2f*>(aRowL + kb);
        v2f aR = *reinterpret_cast<const v2f*>(aRowR + kb);
#pragma unroll
        for (int t = 0; t < 4; ++t) {
            const int n = t * 16 + lrow;
            v2f bL = *reinterpret_cast<const v2f*>(sWl + n * FDIM + kb + half * 2);
            v2f bR = *reinterpret_cast<const v2f*>(sWr + n * FDIM + kb + half * 2);
            acc[t] = __builtin_amdgcn_wmma_f32_16x16x4_f32(
                false, aL, false, bL, (short)0, acc[t], false, false);
            acc[t] = __builtin_amdgcn_wmma_f32_16x16x4_f32(
                false, aR, false, bR, (short)0, acc[t], false, false);
        }
    }

    // Epilogue: C/D layout — VGPR v holds (M = v + half*8, N = n0 + lrow).
#pragma unroll
    for (int t = 0; t < 4; ++t) {
        const int n = t * 16 + lrow;
        const float bv = bias[n];
#pragma unroll
        for (int v = 0; v < 8; ++v) {
            const int m = v + half * 8;
            float val = acc[t][v] + bv;
            if (applyTanh) val = tanhf(val);
            out[(r0 + m) * FDIM + n] = val;
        }
    }
}

// ---------------------------------------------------------------------------
// Host: 4 layers. Scratch: agg + hA in d_ws (51.2MB); d_out doubles as the
// pong buffer; final layer (no tanh) lands in d_out.
// ---------------------------------------------------------------------------
extern "C" void kernel_launch(void* const* d_in, const int* in_sizes, int n_in,
                              void* d_out, int out_size, void* d_ws, size_t ws_size,
                              hipStream_t stream)
{
    const float* x      = (const float*)d_in[0];
    const int*   ei     = (const int*)d_in[1];
    const int*   src    = ei;            // edge_index[0]
    const int*   dst    = ei + NEDGES;   // edge_index[1]
    const float* Wl_in  = (const float*)d_in[2];
    const float* bl_in  = (const float*)d_in[3];
    const float* Wr_in  = (const float*)d_in[4];
    const float* Wl_med = (const float*)d_in[5];
    const float* bl_med = (const float*)d_in[6];
    const float* Wr_med = (const float*)d_in[7];
    const float* Wl_out = (const float*)d_in[8];
    const float* bl_out = (const float*)d_in[9];
    const float* Wr_out = (const float*)d_in[10];

    float* agg = (float*)d_ws;
    float* hA  = agg + (size_t)NNODES * FDIM;
    float* hB  = (float*)d_out;          // ping-pong through the output buffer

    const size_t aggBytes = (size_t)NNODES * FDIM * sizeof(float);

    const dim3 sBlk(256);
    const dim3 sGrid((unsigned)(((size_t)NEDGES * 16 + 255) / 256));
    const dim3 gBlk(256);
    const dim3 gGrid((ROWTILES + 7) / 8);

    auto layer = [&](const float* hin, float* hout, const float* Wl,
                     const float* bl, const float* Wr, int doTanh) {
        hipMemsetAsync(agg, 0, aggBytes, stream);
        scatter_add_kernel<<<sGrid, sBlk, 0, stream>>>(hin, src, dst, agg, NEDGES);
        sage_gemm_kernel<<<gGrid, gBlk, 0, stream>>>(agg, hin, Wl, bl, Wr, hout, doTanh);
    };

    layer(x,  hA,           Wl_in,  bl_in,  Wr_in,  1);  // conv_in + tanh
    layer(hA, hB,           Wl_med, bl_med, Wr_med, 1);  // conv_med #1 + tanh
    layer(hB, hA,           Wl_med, bl_med, Wr_med, 1);  // conv_med #2 + tanh
    layer(hA, (float*)d_out, Wl_out, bl_out, Wr_out, 0); // conv_out (no act)
}